// InfoNCELoss_37383395344507
// MI455X (gfx1250) — compile-verified
//
#include <hip/hip_runtime.h>
#include <hip/hip_bf16.h>
#include <math.h>

typedef __attribute__((ext_vector_type(2))) float v2f;
typedef __attribute__((ext_vector_type(8))) float v8f;

#define TAU_INV   10.0f
#define SHIFT     10.0f      // fixed stabilizer: s = sim/tau <= 10, result invariant to shift
#define PT_THRES  0.9f
#define NCLUST    64
#define EPSN      1e-8f

// ---------------------------------------------------------------------------
// K1: per-row inverse norms  invn[n] = 1/max(||x_n||, eps); zero accumulators.
// x fits in L2 (25.6MB << 192MB); this pass also warms it.
// ---------------------------------------------------------------------------
__global__ void k1_init_norms(const float* __restrict__ x, int N,
                              float* __restrict__ invn, float* __restrict__ denom,
                              float* __restrict__ sums, float* __restrict__ counts) {
  int n = blockIdx.x * blockDim.x + threadIdx.x;
  if (n < NCLUST) { sums[n] = 0.f; counts[n] = 0.f; }
  if (n >= N) return;
  const float4* row = (const float4*)(x + (size_t)n * 64);
  float acc = 0.f;
#pragma unroll
  for (int c = 0; c < 16; ++c) {
    float4 v = row[c];
    acc += v.x * v.x + v.y * v.y + v.z * v.z + v.w * v.w;
  }
  invn[n]  = 1.0f / fmaxf(sqrtf(acc), EPSN);
  denom[n] = 0.f;
}

// ---------------------------------------------------------------------------
// K2: one wave = 16 pairs. Batched 64-dim dot products via chained
// V_WMMA_F32_16X16X4_F32 (A = x[i] rows, B = x[j] cols, diagonal of D = dots).
// A 16x4 f32 layout: lane m (m<16): {K0,K1} in v0,v1 ; lane m+16: {K2,K3}.
// B 4x16 f32 layout: lane n: {K0,K1} ; lane n+16: {K2,K3}. Same gather shape:
//   addr = row*64 + 4*chunk + 2*(lane>>4), 8-byte load.
// D 16x16 f32: diag (k,k) lives at VGPR k, lane k (k<8) and VGPR k-8,
//   lane k+16 (k>=8)  => lanes 0-7 and 24-31, VGPR index = lane&7.
// Stores e = exp(s-10) with sign bit = pos_mask; accumulates neg denominator.
// ---------------------------------------------------------------------------
__global__ void k2_pairs_wmma(const float* __restrict__ x,
                              const int* __restrict__ pi, const int* __restrict__ pj,
                              const int* __restrict__ cid,
                              const unsigned char* __restrict__ recons,
                              const float* __restrict__ pts,
                              const float* __restrict__ invn,
                              float* __restrict__ esigned,
                              float* __restrict__ denom,
                              int P) {
  const int lane = threadIdx.x & 31;
  const int wave = (blockIdx.x * blockDim.x + threadIdx.x) >> 5;
  const int base = wave * 16;
  if (base >= P) return;                       // wave-uniform exit, EXEC stays full

  int pr  = base + (lane & 15);
  int prc = pr < P ? pr : (P - 1);             // clamp tail, no divergence
  int ii  = pi[prc];
  int jj  = pj[prc];
  int ksel = (lane >> 4) << 1;                 // 0 for lanes 0-15, 2 for 16-31
  const float* xi = x + (size_t)ii * 64 + ksel;
  const float* xj = x + (size_t)jj * 64 + ksel;

  v8f acc = {};
#pragma unroll
  for (int c = 0; c < 16; ++c) {               // K = 64 in chunks of 4
    v2f a = *(const v2f*)(xi + c * 4);
    v2f b = *(const v2f*)(xj + c * 4);
    acc = __builtin_amdgcn_wmma_f32_16x16x4_f32(
        /*neg_a=*/false, a, /*neg_b=*/false, b,
        /*c_mod=*/(short)0, acc, /*reuse_a=*/false, /*reuse_b=*/false);
  }

  // diagonal mux: VGPR index = lane&7 for the two diagonal lane groups
  const int k7 = lane & 7;
  float m0 = (k7 & 1) ? acc[1] : acc[0];
  float m1 = (k7 & 1) ? acc[3] : acc[2];
  float m2 = (k7 & 1) ? acc[5] : acc[4];
  float m3 = (k7 & 1) ? acc[7] : acc[6];
  float n0 = (k7 & 2) ? m1 : m0;
  float n1 = (k7 & 2) ? m3 : m2;
  float dsel = (k7 & 4) ? n1 : n0;

  const bool diag = (lane < 8) || (lane >= 24);
  if (diag && pr < P) {
    float s = dsel * invn[ii] * invn[jj] * TAU_INV;
    float e = expf(s - SHIFT);                 // in [e^-20, 1], fp32-safe
    bool extra = (pts[ii] > PT_THRES) && (pts[jj] > PT_THRES) &&
                 (recons[ii] != 0) && (recons[jj] != 0);
    bool pos = (cid[ii] == cid[jj]) && extra;
    esigned[pr] = pos ? -e : e;                // sign bit encodes pos_mask (e>0 always)
    if (!pos) atomicAdd(&denom[ii], e);        // segment_sum of negative pairs by anchor
  }
}

// ---------------------------------------------------------------------------
// K3: positive pairs only (~0.5%): loss = log(e + denom_i) - log(e),
// accumulate per-cluster sum & count.
// ---------------------------------------------------------------------------
__global__ void k3_loss(const float* __restrict__ esigned,
                        const int* __restrict__ pi,
                        const int* __restrict__ cid,
                        const float* __restrict__ denom,
                        float* __restrict__ sums, float* __restrict__ counts,
                        int P) {
  int p = blockIdx.x * blockDim.x + threadIdx.x;
  if (p >= P) return;
  float v = esigned[p];
  if (!(v < 0.f)) return;                      // only pos pairs (encoded negative)
  float e = -v;
  int ii = pi[p];
  float d = denom[ii];
  float loss = logf(e + d) - logf(e);
  int lab = cid[ii];
  atomicAdd(&sums[lab], loss);
  atomicAdd(&counts[lab], 1.0f);
}

// ---------------------------------------------------------------------------
// K4: mean over positives per cluster, then mean over present clusters.
// ---------------------------------------------------------------------------
__global__ void k4_final(const float* __restrict__ sums,
                         const float* __restrict__ counts,
                         float* __restrict__ out) {
  __shared__ float sm[NCLUST];
  __shared__ float sp[NCLUST];
  int t = threadIdx.x;
  float c = counts[t];
  bool present = c > 0.f;
  sm[t] = present ? sums[t] / fmaxf(c, 1.0f) : 0.f;
  sp[t] = present ? 1.f : 0.f;
  __syncthreads();
  if (t == 0) {
    float a = 0.f, b = 0.f;
    for (int q = 0; q < NCLUST; ++q) { a += sm[q]; b += sp[q]; }
    out[0] = a / fmaxf(b, 1.0f);
  }
}

extern "C" void kernel_launch(void* const* d_in, const int* in_sizes, int n_in,
                              void* d_out, int out_size, void* d_ws, size_t ws_size,
                              hipStream_t stream) {
  const float*         x      = (const float*)d_in[0];
  const int*           pp     = (const int*)d_in[1];
  const int*           cid    = (const int*)d_in[2];
  const unsigned char* recons = (const unsigned char*)d_in[3];
  const float*         pts    = (const float*)d_in[4];

  const int N = in_sizes[2];
  const int P = in_sizes[1] / 2;
  const int* pi = pp;
  const int* pj = pp + P;

  // workspace layout (floats): [esigned P][denom N][invn N][sums 64][counts 64]
  float* ws      = (float*)d_ws;
  float* esigned = ws;
  float* denom   = ws + (size_t)P;
  float* invn    = ws + (size_t)P + N;
  float* sums    = ws + (size_t)P + 2 * (size_t)N;
  float* counts  = sums + NCLUST;

  {
    int threads = 256;
    int blocks  = (N + threads - 1) / threads;
    hipLaunchKernelGGL(k1_init_norms, dim3(blocks), dim3(threads), 0, stream,
                       x, N, invn, denom, sums, counts);
  }
  {
    int tiles  = (P + 15) / 16;      // 16 pairs per wave
    int wpb    = 8;                  // 8 waves (256 threads) per block
    int blocks = (tiles + wpb - 1) / wpb;
    hipLaunchKernelGGL(k2_pairs_wmma, dim3(blocks), dim3(wpb * 32), 0, stream,
                       x, pi, pj, cid, recons, pts, invn, esigned, denom, P);
  }
  {
    int threads = 256;
    int blocks  = (P + threads - 1) / threads;
    hipLaunchKernelGGL(k3_loss, dim3(blocks), dim3(threads), 0, stream,
                       esigned, pi, cid, denom, sums, counts, P);
  }
  hipLaunchKernelGGL(k4_final, dim3(1), dim3(NCLUST), 0, stream,
                     sums, counts, (float*)d_out);
}